// NexusV2_8366596292757
// MI455X (gfx1250) — compile-verified
//
#include <hip/hip_runtime.h>
#include <math.h>

// ---------------------------------------------------------------------------
// CentroidAddressableManifold.read — MI455X (gfx1250, wave32)
//
// B*T = 4096 tokens, D = 1024, 32 slots/bucket, 512 buckets.
// Bandwidth-bound (~160MB unique working set, fits 192MB L2; ~0.55 GFLOP).
// fp32 throughout (exact vs reference). probs @ b_vals uses chained
// V_WMMA_F32_16X16X4_F32 (fp32-exact matrix pipe).
// ---------------------------------------------------------------------------

#define DIM 1024
#define NBUCKETS 512
#define SLOTS 32
#define TAU 0.1f
#define BLOCK 128          // 4 waves of 32
#define QPT (DIM / BLOCK)  // 8 elements per thread

typedef __attribute__((ext_vector_type(2))) float v2f;
typedef __attribute__((ext_vector_type(8))) float v8f;

__device__ __forceinline__ float wave_sum(float v) {
#pragma unroll
    for (int o = 16; o > 0; o >>= 1) v += __shfl_xor(v, o, 32);
    return v;
}
__device__ __forceinline__ float wave_max(float v) {
#pragma unroll
    for (int o = 16; o > 0; o >>= 1) v = fmaxf(v, __shfl_xor(v, o, 32));
    return v;
}

__global__ __launch_bounds__(BLOCK) void cam_read_kernel(
    const float* __restrict__ query_emb,   // [nTok, D]
    const int*   __restrict__ tids,        // [nTok]
    const float* __restrict__ slot_keys,   // [NBUCKETS*SLOTS, D]
    const float* __restrict__ slot_values, // [NBUCKETS*SLOTS, D]
    const int*   __restrict__ slot_tids,   // [NBUCKETS*SLOTS]
    const float* __restrict__ centroids,   // [NBUCKETS, D]
    float*       __restrict__ out)         // [nTok, D]
{
    __shared__ float s_uq[DIM];      // unified query (fp32)
    __shared__ float s_scores[SLOTS];
    __shared__ float s_probs[SLOTS];
    __shared__ float s_red[BLOCK / 32];

    const int t      = blockIdx.x;
    const int tidx   = threadIdx.x;
    const int lane   = tidx & 31;
    const int wave   = tidx >> 5;

    const int tok_id = tids[t];
    const int bucket = ((unsigned)tok_id) & (NBUCKETS - 1); // tok_id >= 0

    const float* qrow  = query_emb  + (size_t)t * DIM;
    const float* crow  = centroids  + (size_t)bucket * DIM;
    const float* kbase = slot_keys  + (size_t)bucket * SLOTS * DIM;
    const float* vbase = slot_values + (size_t)bucket * SLOTS * DIM;

    // ---- Phase 1: unified query = l2norm(0.5*l2norm(q) + 0.5*centroid) ----
    float qv[QPT];
    float ss = 0.f;
#pragma unroll
    for (int j = 0; j < QPT; ++j) {
        qv[j] = qrow[tidx + j * BLOCK];     // coalesced, stride-BLOCK
        ss += qv[j] * qv[j];
    }
    ss = wave_sum(ss);
    if (lane == 0) s_red[wave] = ss;
    __syncthreads();
    ss = s_red[0] + s_red[1] + s_red[2] + s_red[3];
    const float invq = 1.f / fmaxf(sqrtf(ss), 1e-12f);

    float uv[QPT];
    float ss2 = 0.f;
#pragma unroll
    for (int j = 0; j < QPT; ++j) {
        float u = 0.5f * qv[j] * invq + 0.5f * crow[tidx + j * BLOCK];
        uv[j] = u;
        ss2 += u * u;
    }
    ss2 = wave_sum(ss2);
    __syncthreads();                        // s_red reuse
    if (lane == 0) s_red[wave] = ss2;
    __syncthreads();
    ss2 = s_red[0] + s_red[1] + s_red[2] + s_red[3];
    const float invu = 1.f / fmaxf(sqrtf(ss2), 1e-12f);
#pragma unroll
    for (int j = 0; j < QPT; ++j) s_uq[tidx + j * BLOCK] = uv[j] * invu;

    // Warm L2/L0 with this token's value rows while scores are computed.
    // 32 slots * 4KB = 128KB = 1024 x 128B lines; 8 lines per thread.
#pragma unroll
    for (int k = 0; k < 8; ++k)
        __builtin_prefetch(vbase + ((size_t)tidx + (size_t)k * BLOCK) * 32, 0, 1);

    __syncthreads();

    // ---- Phase 2: scores[s] = dot(uq, keys[s]); wave w owns slots 8w..8w+7 -
    const float4* urow4 = (const float4*)s_uq;
#pragma unroll 1
    for (int si = 0; si < 8; ++si) {
        const int s = wave * 8 + si;
        const float4* krow4 = (const float4*)(kbase + (size_t)s * DIM);
        float p = 0.f;
#pragma unroll
        for (int j = 0; j < DIM / (32 * 4); ++j) {   // 8 iters
            float4 k4 = krow4[j * 32 + lane];        // 512B coalesced / instr
            float4 u4 = urow4[j * 32 + lane];        // LDS b128
            p += k4.x * u4.x + k4.y * u4.y + k4.z * u4.z + k4.w * u4.w;
        }
        p = wave_sum(p);
        if (lane == 0) s_scores[s] = p;
    }
    __syncthreads();

    // ---- Phase 3: hard-match vs softmax probs (wave 0, 32 lanes = 32 slots) -
    if (tidx < 32) {
        const int   s    = tidx;
        const float sc   = s_scores[s];
        const int   stid = slot_tids[bucket * SLOTS + s];
        const float m    = (stid == tok_id) ? 1.f : 0.f;
        const float msum = wave_sum(m);
        const float x    = sc * (1.f / TAU);
        const float mx   = wave_max(x);
        const float e    = __expf(x - mx);
        const float esum = wave_sum(e);
        s_probs[s] = (msum > 0.f) ? (m / (msum + 1e-9f)) : (e / esum);
    }
    __syncthreads();

    // ---- Phase 4: val[d] = sum_s probs[s] * vals[s][d] via fp32 WMMA -------
    // Chain 8x V_WMMA_F32_16X16X4_F32 over K=32 slots per 16-wide d-chunk.
    // A (16x4, rows replicated with probs): lane l holds K = 2*(l/16)+v.
    // B (4x16): lane l = column l%16, VGPR v holds row K = 2*(l/16)+v.
    // D (16x16): all rows equal; VGPR0 lanes 0-15 carry (M=0, N=lane).
    const int kh = lane >> 4;   // K half select
    const int n  = lane & 15;   // output column within chunk

    v2f a[8];
#pragma unroll
    for (int j = 0; j < 8; ++j) {
        a[j].x = s_probs[4 * j + kh * 2 + 0];
        a[j].y = s_probs[4 * j + kh * 2 + 1];
    }

    float* orow = out + (size_t)t * DIM;
#pragma unroll 1
    for (int c = wave; c < DIM / 16; c += BLOCK / 32) {  // 16 chunks per wave
        const int d0 = c * 16;
        v8f acc = {};
#pragma unroll
        for (int j = 0; j < 8; ++j) {
            const int r = 4 * j + kh * 2;
            v2f b;
            b.x = vbase[(size_t)r       * DIM + d0 + n];   // 64B/row coalesced
            b.y = vbase[(size_t)(r + 1) * DIM + d0 + n];
            acc = __builtin_amdgcn_wmma_f32_16x16x4_f32(
                false, a[j], false, b, (short)0, acc, false, false);
        }
        if (lane < 16) orow[d0 + lane] = acc[0];
    }
}

extern "C" void kernel_launch(void* const* d_in, const int* in_sizes, int n_in,
                              void* d_out, int out_size, void* d_ws, size_t ws_size,
                              hipStream_t stream) {
    const float* query_emb   = (const float*)d_in[0];
    const int*   tids        = (const int*)  d_in[1];
    const float* slot_keys   = (const float*)d_in[2];
    const float* slot_values = (const float*)d_in[3];
    const int*   slot_tids   = (const int*)  d_in[4];
    const float* centroids   = (const float*)d_in[5];
    float*       out         = (float*)d_out;

    const int nTok = in_sizes[1];   // B*T
    if (nTok <= 0) return;

    hipLaunchKernelGGL(cam_read_kernel, dim3(nTok), dim3(BLOCK), 0, stream,
                       query_emb, tids, slot_keys, slot_values, slot_tids,
                       centroids, out);
}